// GLASSConv_45397804318728
// MI455X (gfx1250) — compile-verified
//
#include <hip/hip_runtime.h>
#include <hip/hip_bf16.h>
#include <stdint.h>

// ---------------------------------------------------------------------------
// GLASS-GNN forward for MI455X (gfx1250, wave32, WMMA).
// GEMMs use V_WMMA_F32_16X16X4_F32 (exact f32, matrix pipe). A-tiles are
// staged into LDS with the CDNA5 async global->LDS datapath
// (GLOBAL_LOAD_ASYNC_TO_LDS_B128 + ASYNCcnt), bypassing VGPRs. Edge softmax
// aggregation uses L2-resident f32/u32 atomics. All 256-thread blocks = 8
// wave32 waves; each wave computes one 16x16 f32 output tile.
// ---------------------------------------------------------------------------

#define N_NODES   100000
#define N_EDGES   1600000
#define DI        128
#define ZMIX      0.8f
#define EPS_GN    1e-5f
#define NEG_SLOPE 0.2f

typedef __attribute__((ext_vector_type(2))) float v2f;
typedef __attribute__((ext_vector_type(8))) float v8f;

// Async copy of 16 bytes global -> LDS (CDNA5 direct datapath, ASYNCcnt).
__device__ __forceinline__ void async_g2l_b128(unsigned lds_off, const void* g) {
  unsigned long long ga = (unsigned long long)(uintptr_t)g;
  asm volatile("global_load_async_to_lds_b128 %0, %1, off"
               :: "v"(lds_off), "v"(ga) : "memory");
}
__device__ __forceinline__ void wait_async0() {
  asm volatile("s_wait_asynccnt 0x0" ::: "memory");
}
__device__ __forceinline__ unsigned lds_off_of(const void* p) {
  return (unsigned)(uintptr_t)p;   // low 32 bits of flat shared addr = LDS offset
}

// Order-preserving float<->uint mapping so we can use u32 atomicMax for
// the segment-max over (possibly negative) attention logits.
__device__ __forceinline__ unsigned f32_to_ord(float f) {
  unsigned u = __float_as_uint(f);
  return (u & 0x80000000u) ? ~u : (u | 0x80000000u);
}
__device__ __forceinline__ float ord_to_f32(unsigned k) {
  unsigned u = (k & 0x80000000u) ? (k & 0x7fffffffu) : ~k;
  return __uint_as_float(u);
}

// ---------------------------------------------------------------------------
// K0: zero the 256-float column-stats accumulator (sum | sumsq).
// ---------------------------------------------------------------------------
__global__ __launch_bounds__(256) void k_zero_accum(float* __restrict__ accum) {
  accum[threadIdx.x] = 0.0f;
}

// ---------------------------------------------------------------------------
// K1: x = mix(relu(x_ @ W_t1^T + b_t1), relu(x_ @ W_t0^T + b_t0), mask)
// One block = 16 rows x 128 cols. A tile async-staged in LDS (stride 132
// dwords: 16B-aligned b128 targets, 8B-aligned + bank-spread b64 fragment
// reads). Each wave: 16x16 tile, 32 K-steps x 2 weights = 64 WMMAs.
// ---------------------------------------------------------------------------
__global__ __launch_bounds__(256) void k_transform_mix(
    const float* __restrict__ x_,
    const float* __restrict__ Wt0, const float* __restrict__ bt0,
    const float* __restrict__ Wt1, const float* __restrict__ bt1,
    const unsigned char* __restrict__ mask,
    float* __restrict__ xout)
{
  __shared__ float lA[16 * 132];
  const int tid = threadIdx.x;
  const int rowBase = blockIdx.x * 16;

  // Stage 16x128 f32 A tile: 2048 floats, 8 per thread (2x async b128).
  {
    int idx = tid * 8;
    int r = idx >> 7;
    int c = idx & 127;
    const float* gsrc = x_ + (size_t)(rowBase + r) * DI + c;
    unsigned dst = lds_off_of(&lA[r * 132 + c]);
    async_g2l_b128(dst,       gsrc);
    async_g2l_b128(dst + 16u, gsrc + 4);
    wait_async0();
  }
  __syncthreads();

  const int lane  = tid & 31;
  const int wave  = tid >> 5;
  const int mrow  = lane & 15;        // A-fragment row (M)
  const int hi    = lane >> 4;        // K-pair selector (lanes 16-31 -> K+2)
  const int nglob = wave * 16 + mrow; // output column == weight row

  const float* a_row = &lA[mrow * 132];
  const float* w0row = Wt0 + (size_t)nglob * DI;
  const float* w1row = Wt1 + (size_t)nglob * DI;
  __builtin_prefetch(w0row, 0, 1);
  __builtin_prefetch(w1row, 0, 1);

  v8f acc0 = {};
  v8f acc1 = {};
  for (int kk = 0; kk < DI; kk += 4) {
    int ko = kk + (hi << 1);
    v2f a  = *(const v2f*)(a_row + ko);
    v2f b0 = *(const v2f*)(w0row + ko);
    v2f b1 = *(const v2f*)(w1row + ko);
    acc0 = __builtin_amdgcn_wmma_f32_16x16x4_f32(false, a, false, b0,
                                                 (short)0, acc0, false, false);
    acc1 = __builtin_amdgcn_wmma_f32_16x16x4_f32(false, a, false, b1,
                                                 (short)0, acc1, false, false);
  }

  const float bb0 = bt0[nglob];
  const float bb1 = bt1[nglob];
  const int col = nglob;
#pragma unroll
  for (int j = 0; j < 8; ++j) {
    int row = rowBase + j + (hi << 3);   // C layout: M = j + 8*(lane>=16)
    float r0v = fmaxf(acc0[j] + bb0, 0.0f);
    float r1v = fmaxf(acc1[j] + bb1, 0.0f);
    float y = mask[row] ? (ZMIX * r1v + (1.0f - ZMIX) * r0v)
                        : (ZMIX * r0v + (1.0f - ZMIX) * r1v);
    xout[(size_t)row * DI + col] = y;
  }
}

// ---------------------------------------------------------------------------
// K2: xp = x @ gat_W^T   (same tiling, single weight, no bias/activation)
// ---------------------------------------------------------------------------
__global__ __launch_bounds__(256) void k_gat_proj(
    const float* __restrict__ xin,
    const float* __restrict__ gatW,
    float* __restrict__ xp)
{
  __shared__ float lA[16 * 132];
  const int tid = threadIdx.x;
  const int rowBase = blockIdx.x * 16;
  {
    int idx = tid * 8;
    int r = idx >> 7;
    int c = idx & 127;
    const float* gsrc = xin + (size_t)(rowBase + r) * DI + c;
    unsigned dst = lds_off_of(&lA[r * 132 + c]);
    async_g2l_b128(dst,       gsrc);
    async_g2l_b128(dst + 16u, gsrc + 4);
    wait_async0();
  }
  __syncthreads();

  const int lane  = tid & 31;
  const int wave  = tid >> 5;
  const int mrow  = lane & 15;
  const int hi    = lane >> 4;
  const int nglob = wave * 16 + mrow;

  const float* a_row = &lA[mrow * 132];
  const float* wrow  = gatW + (size_t)nglob * DI;
  __builtin_prefetch(wrow, 0, 1);

  v8f acc = {};
  for (int kk = 0; kk < DI; kk += 4) {
    int ko = kk + (hi << 1);
    v2f a = *(const v2f*)(a_row + ko);
    v2f b = *(const v2f*)(wrow + ko);
    acc = __builtin_amdgcn_wmma_f32_16x16x4_f32(false, a, false, b,
                                                (short)0, acc, false, false);
  }
  const int col = nglob;
#pragma unroll
  for (int j = 0; j < 8; ++j) {
    int row = rowBase + j + (hi << 3);
    xp[(size_t)row * DI + col] = acc[j];
  }
}

// ---------------------------------------------------------------------------
// K3: per node: a_src = xp . att_src, a_dst = xp . att_dst (wave32 butterfly
// reduction), plus init amax_key=0 (ordered -inf), denom=0, out_accum row=0.
// One wave per node; 8 nodes per block.
// ---------------------------------------------------------------------------
__global__ __launch_bounds__(256) void k_attn_scalars(
    const float* __restrict__ xp,
    const float* __restrict__ att_src, const float* __restrict__ att_dst,
    float* __restrict__ a_src, float* __restrict__ a_dst,
    unsigned* __restrict__ amax_key, float* __restrict__ denom,
    float* __restrict__ out_accum)
{
  int node = blockIdx.x * 8 + (threadIdx.x >> 5);
  if (node >= N_NODES) return;
  int lane = threadIdx.x & 31;
  int c = lane * 4;

  float4 v  = *(const float4*)(xp + (size_t)node * DI + c);
  float4 as = *(const float4*)(att_src + c);
  float4 ad = *(const float4*)(att_dst + c);
  float ps = v.x * as.x + v.y * as.y + v.z * as.z + v.w * as.w;
  float pd = v.x * ad.x + v.y * ad.y + v.z * ad.z + v.w * ad.w;
#pragma unroll
  for (int o = 16; o > 0; o >>= 1) {
    ps += __shfl_xor(ps, o, 32);
    pd += __shfl_xor(pd, o, 32);
  }
  float4 z4 = {0.0f, 0.0f, 0.0f, 0.0f};
  *(float4*)(out_accum + (size_t)node * DI + c) = z4;
  if (lane == 0) {
    a_src[node] = ps;
    a_dst[node] = pd;
    amax_key[node] = 0u;   // ordered-uint minimum
    denom[node] = 0.0f;
  }
}

// ---------------------------------------------------------------------------
// K4: edge sweep 1 — segment max of leaky_relu(a_src[s]+a_dst[d]) per dst.
// Edges [0,E) from edge_index (int64), [E, E+N) are self loops.
// ---------------------------------------------------------------------------
__global__ __launch_bounds__(256) void k_edge_max(
    const long long* __restrict__ ei,
    const float* __restrict__ a_src, const float* __restrict__ a_dst,
    unsigned* __restrict__ amax_key)
{
  long long e = (long long)blockIdx.x * 256 + threadIdx.x;
  if (e >= (long long)N_EDGES + N_NODES) return;
  int s, d;
  if (e < N_EDGES) { s = (int)ei[e]; d = (int)ei[N_EDGES + e]; }
  else             { s = d = (int)(e - N_EDGES); }
  float a = a_src[s] + a_dst[d];
  a = (a > 0.0f) ? a : NEG_SLOPE * a;
  atomicMax(&amax_key[d], f32_to_ord(a));
}

// ---------------------------------------------------------------------------
// K5: edge sweep 2 — ex[e] = exp(alpha - amax[dst]); denom[dst] += ex.
// ---------------------------------------------------------------------------
__global__ __launch_bounds__(256) void k_edge_exp(
    const long long* __restrict__ ei,
    const float* __restrict__ a_src, const float* __restrict__ a_dst,
    const unsigned* __restrict__ amax_key,
    float* __restrict__ ex, float* __restrict__ denom)
{
  long long e = (long long)blockIdx.x * 256 + threadIdx.x;
  if (e >= (long long)N_EDGES + N_NODES) return;
  int s, d;
  if (e < N_EDGES) { s = (int)ei[e]; d = (int)ei[N_EDGES + e]; }
  else             { s = d = (int)(e - N_EDGES); }
  float a = a_src[s] + a_dst[d];
  a = (a > 0.0f) ? a : NEG_SLOPE * a;
  float exv = expf(a - ord_to_f32(amax_key[d]));
  ex[e] = exv;
  atomicAdd(&denom[d], exv);
}

// ---------------------------------------------------------------------------
// K6: edge sweep 3 — out[dst] += xp[src] * (ex / (denom[dst]+1e-16)).
// One wave per edge, each lane handles 4 columns (b128 gather + 4 f32 atomics).
// ---------------------------------------------------------------------------
__global__ __launch_bounds__(256) void k_edge_scatter(
    const long long* __restrict__ ei,
    const float* __restrict__ ex, const float* __restrict__ denom,
    const float* __restrict__ xp, float* __restrict__ out_accum)
{
  long long e = (long long)blockIdx.x * 8 + (threadIdx.x >> 5);
  if (e >= (long long)N_EDGES + N_NODES) return;
  int lane = threadIdx.x & 31;
  int s, d;
  if (e < N_EDGES) { s = (int)ei[e]; d = (int)ei[N_EDGES + e]; }
  else             { s = d = (int)(e - N_EDGES); }
  float coef = ex[e] / (denom[d] + 1e-16f);
  int c = lane * 4;
  float4 v = *(const float4*)(xp + (size_t)s * DI + c);
  float* o = out_accum + (size_t)d * DI + c;
  atomicAdd(o + 0, v.x * coef);
  atomicAdd(o + 1, v.y * coef);
  atomicAdd(o + 2, v.z * coef);
  atomicAdd(o + 3, v.w * coef);
}

// ---------------------------------------------------------------------------
// K7: per-column sum / sum-of-squares of (out + gat_bias) over all nodes.
// thread = column (coalesced row reads), block = 128-row slab, then atomics.
// ---------------------------------------------------------------------------
__global__ __launch_bounds__(128) void k_colstats(
    const float* __restrict__ out_accum,
    const float* __restrict__ gat_bias,
    float* __restrict__ accum)
{
  int c = threadIdx.x;
  float gb = gat_bias[c];
  float s = 0.0f, s2 = 0.0f;
  int r0 = blockIdx.x * 128;
  int r1 = (r0 + 128 < N_NODES) ? (r0 + 128) : N_NODES;
  for (int r = r0; r < r1; ++r) {
    float o = out_accum[(size_t)r * DI + c] + gb;
    s  += o;
    s2 += o * o;
  }
  atomicAdd(&accum[c], s);
  atomicAdd(&accum[DI + c], s2);
}

// ---------------------------------------------------------------------------
// K8: finale — graph-norm, concat [norm(out) || x_] (K=256), two GEMMs
// (W_c1, W_c0) via WMMA, bias, mask-mix, store to d_out.
// var = E[o^2] - (2g - g^2) mu^2  with  cen = o - g*mu,  mu = E[o].
// Norm half is computed through VGPRs; raw x_ half uses async global->LDS.
// ---------------------------------------------------------------------------
__global__ __launch_bounds__(256) void k_final(
    const float* __restrict__ out_accum, const float* __restrict__ x_,
    const float* __restrict__ accum,
    const float* __restrict__ gat_bias,
    const float* __restrict__ gn_w, const float* __restrict__ gn_b,
    const float* __restrict__ gn_ms,
    const float* __restrict__ Wc0, const float* __restrict__ bc0,
    const float* __restrict__ Wc1, const float* __restrict__ bc1,
    const unsigned char* __restrict__ mask,
    float* __restrict__ y)
{
  __shared__ float lA[16 * 260];
  __shared__ float lmean[DI];   // g * mu
  __shared__ float lscale[DI];  // invstd * gn_weight
  __shared__ float lshift[DI];  // gn_bias

  const int tid = threadIdx.x;
  if (tid < DI) {
    const float invN = 1.0f / (float)N_NODES;
    float mu = accum[tid] * invN;
    float e2 = accum[DI + tid] * invN;
    float g  = gn_ms[tid];
    float var = e2 - (2.0f * g - g * g) * mu * mu;
    float inv = rsqrtf(var + EPS_GN);
    lmean[tid]  = g * mu;
    lscale[tid] = inv * gn_w[tid];
    lshift[tid] = gn_b[tid];
  }
  __syncthreads();

  const int rowBase = blockIdx.x * 16;
  // Stage 16x256 xc tile: 4096 floats, 4x float4 per thread.
  // Columns [0,128): graph-normed out (VGPR math path).
  // Columns [128,256): raw x_ copy (async global->LDS path).
#pragma unroll
  for (int t = 0; t < 4; ++t) {
    int idx = (t * 256 + tid) * 4;
    int r = idx >> 8;
    int c = idx & 255;
    if (c < DI) {
      float4 o = *(const float4*)(out_accum + (size_t)(rowBase + r) * DI + c);
      float4 gb = *(const float4*)(gat_bias + c);
      float4 v;
      v.x = (o.x + gb.x - lmean[c + 0]) * lscale[c + 0] + lshift[c + 0];
      v.y = (o.y + gb.y - lmean[c + 1]) * lscale[c + 1] + lshift[c + 1];
      v.z = (o.z + gb.z - lmean[c + 2]) * lscale[c + 2] + lshift[c + 2];
      v.w = (o.w + gb.w - lmean[c + 3]) * lscale[c + 3] + lshift[c + 3];
      *(float4*)&lA[r * 260 + c] = v;
    } else {
      async_g2l_b128(lds_off_of(&lA[r * 260 + c]),
                     x_ + (size_t)(rowBase + r) * DI + (c - DI));
    }
  }
  wait_async0();
  __syncthreads();

  const int lane  = tid & 31;
  const int wave  = tid >> 5;
  const int mrow  = lane & 15;
  const int hi    = lane >> 4;
  const int nglob = wave * 16 + mrow;

  const float* a_row = &lA[mrow * 260];
  const float* w0row = Wc0 + (size_t)nglob * 256;
  const float* w1row = Wc1 + (size_t)nglob * 256;
  __builtin_prefetch(w0row, 0, 1);
  __builtin_prefetch(w1row, 0, 1);

  v8f acc0 = {};
  v8f acc1 = {};
  for (int kk = 0; kk < 256; kk += 4) {
    int ko = kk + (hi << 1);
    v2f a  = *(const v2f*)(a_row + ko);
    v2f b0 = *(const v2f*)(w0row + ko);
    v2f b1 = *(const v2f*)(w1row + ko);
    acc0 = __builtin_amdgcn_wmma_f32_16x16x4_f32(false, a, false, b0,
                                                 (short)0, acc0, false, false);
    acc1 = __builtin_amdgcn_wmma_f32_16x16x4_f32(false, a, false, b1,
                                                 (short)0, acc1, false, false);
  }

  const float bb0 = bc0[nglob];
  const float bb1 = bc1[nglob];
  const int col = nglob;
#pragma unroll
  for (int j = 0; j < 8; ++j) {
    int row = rowBase + j + (hi << 3);
    float y0v = acc0[j] + bb0;
    float y1v = acc1[j] + bb1;
    float r = mask[row] ? (ZMIX * y1v + (1.0f - ZMIX) * y0v)
                        : (ZMIX * y0v + (1.0f - ZMIX) * y1v);
    y[(size_t)row * DI + col] = r;
  }
}

// ---------------------------------------------------------------------------
// Host launcher. Input order per setup_inputs():
//  0:x_ 1:edge_attr 2:W_t0 3:b_t0 4:W_t1 5:b_t1 6:gat_W 7:att_src 8:att_dst
//  9:gat_bias 10:gn_weight 11:gn_bias 12:gn_mean_scale 13:W_c0 14:b_c0
// 15:W_c1 16:b_c1 17:edge_index(int64) 18:mask(bool)
// ---------------------------------------------------------------------------
extern "C" void kernel_launch(void* const* d_in, const int* in_sizes, int n_in,
                              void* d_out, int out_size, void* d_ws, size_t ws_size,
                              hipStream_t stream) {
  (void)in_sizes; (void)n_in; (void)out_size; (void)ws_size;

  const float* x_       = (const float*)d_in[0];
  const float* W_t0     = (const float*)d_in[2];
  const float* b_t0     = (const float*)d_in[3];
  const float* W_t1     = (const float*)d_in[4];
  const float* b_t1     = (const float*)d_in[5];
  const float* gat_W    = (const float*)d_in[6];
  const float* att_src  = (const float*)d_in[7];
  const float* att_dst  = (const float*)d_in[8];
  const float* gat_bias = (const float*)d_in[9];
  const float* gn_w     = (const float*)d_in[10];
  const float* gn_b     = (const float*)d_in[11];
  const float* gn_ms    = (const float*)d_in[12];
  const float* W_c0     = (const float*)d_in[13];
  const float* b_c0     = (const float*)d_in[14];
  const float* W_c1     = (const float*)d_in[15];
  const float* b_c1     = (const float*)d_in[16];
  const long long* ei   = (const long long*)d_in[17];
  const unsigned char* mask = (const unsigned char*)d_in[18];
  float* y = (float*)d_out;

  // Workspace layout (bytes).
  char* ws = (char*)d_ws;
  const size_t SZ_ND = (size_t)N_NODES * DI * sizeof(float);   // 51,200,000
  const size_t SZ_N  = (size_t)N_NODES * sizeof(float);        //    400,000
  float*    x     = (float*)   (ws);
  float*    xp    = (float*)   (ws + SZ_ND);
  float*    outa  = (float*)   (ws + 2 * SZ_ND);
  float*    asrc  = (float*)   (ws + 3 * SZ_ND);
  float*    adst  = (float*)   (ws + 3 * SZ_ND + SZ_N);
  unsigned* amaxk = (unsigned*)(ws + 3 * SZ_ND + 2 * SZ_N);
  float*    denom = (float*)   (ws + 3 * SZ_ND + 3 * SZ_N);
  float*    ex    = (float*)   (ws + 3 * SZ_ND + 4 * SZ_N);
  float*    accum = (float*)   (ws + 3 * SZ_ND + 4 * SZ_N
                                   + (size_t)(N_EDGES + N_NODES) * sizeof(float));

  const int rowBlocks  = N_NODES / 16;                 // 6250 (exact)
  const long long ETOT = (long long)N_EDGES + N_NODES; // 1,700,000
  const int edgeBlk1   = (int)((ETOT + 255) / 256);
  const int edgeBlk32  = (int)((ETOT + 7) / 8);        // 1 wave / edge
  const int statBlocks = (N_NODES + 127) / 128;

  k_zero_accum   <<<1,          256, 0, stream>>>(accum);
  k_transform_mix<<<rowBlocks,  256, 0, stream>>>(x_, W_t0, b_t0, W_t1, b_t1, mask, x);
  k_gat_proj     <<<rowBlocks,  256, 0, stream>>>(x, gat_W, xp);
  k_attn_scalars <<<(N_NODES + 7) / 8, 256, 0, stream>>>(xp, att_src, att_dst,
                                                         asrc, adst, amaxk, denom, outa);
  k_edge_max     <<<edgeBlk1,   256, 0, stream>>>(ei, asrc, adst, amaxk);
  k_edge_exp     <<<edgeBlk1,   256, 0, stream>>>(ei, asrc, adst, amaxk, ex, denom);
  k_edge_scatter <<<edgeBlk32,  256, 0, stream>>>(ei, ex, denom, xp, outa);
  k_colstats     <<<statBlocks, 128, 0, stream>>>(outa, gat_bias, accum);
  k_final        <<<rowBlocks,  256, 0, stream>>>(outa, x_, accum, gat_bias,
                                                  gn_w, gn_b, gn_ms,
                                                  W_c0, b_c0, W_c1, b_c1, mask, y);
}